// AttentionCell_15625091023560
// MI455X (gfx1250) — compile-verified
//
#include <hip/hip_runtime.h>
#include <math.h>

typedef __attribute__((ext_vector_type(2))) float v2f;
typedef __attribute__((ext_vector_type(8))) float v8f;

#define DIM   512
#define TLEN  1024
#define BATCH 8
#define NROWS (BATCH * TLEN)   // 8192
#define WIN   16
#define LEFTW 16

// ---------------------------------------------------------------------------
// Kernel 1: Q = X@M, K = X@C, V = X@Vw via V_WMMA_F32_16X16X4_F32.
// One wave computes a 16(M) x 64(N) tile (4 accumulators), K-loop step 4.
// Fragment layouts per CDNA5 ISA 7.12.2:
//   A (16x4):  lane m (0..15): v2f = A[m][k], A[m][k+1]; lanes 16..31: K+2,K+3
//   B (4x16):  lane n (0..15): v2f = B[k][n], B[k+1][n]; lanes 16..31: K+2,K+3
//   C/D 16x16: VGPR r -> row r (lanes 0..15) / row r+8 (lanes 16..31), col = lane%16
// ---------------------------------------------------------------------------
__global__ __launch_bounds__(256) void qkv_gemm_wmma(
    const float* __restrict__ X,
    const float* __restrict__ Mw,
    const float* __restrict__ Cw,
    const float* __restrict__ Vw,
    float* __restrict__ Qo,
    float* __restrict__ Ko,
    float* __restrict__ Vo)
{
    const int lane  = threadIdx.x & 31;
    const int gwave = blockIdx.x * (blockDim.x >> 5) + (threadIdx.x >> 5);

    // 4096 tiles per matrix: 512 m-tiles x 8 n-groups(64-wide)
    const int mat  = gwave >> 12;          // 0..2
    const int tile = gwave & 4095;
    const int tm   = tile & 511;           // m-tile
    const int tn   = tile >> 9;            // n-group

    const float* W = (mat == 0) ? Mw : (mat == 1) ? Cw : Vw;
    float*       O = (mat == 0) ? Qo : (mat == 1) ? Ko : Vo;

    const int row0 = tm * 16;
    const int n0   = tn * 64;

    const int am  = lane & 15;             // A row within tile
    const int khi = (lane >> 4) << 1;      // K sub-offset: 0 or 2
    const int bn  = lane & 15;             // B col within 16-group

    v8f acc0 = {}, acc1 = {}, acc2 = {}, acc3 = {};

    const float* xrow = X + (size_t)(row0 + am) * DIM;

    for (int k = 0; k < DIM; k += 4) {
        // A fragment: X[row0+am][k+khi .. k+khi+1]  (contiguous -> b64 load)
        v2f a = *(const v2f*)(xrow + k + khi);

        // B fragments: W[k+khi][n], W[k+khi+1][n] for 4 column groups
        const float* wp = W + (size_t)(k + khi) * DIM + n0 + bn;
        v2f b0 = { wp[ 0], wp[DIM +  0] };
        v2f b1 = { wp[16], wp[DIM + 16] };
        v2f b2 = { wp[32], wp[DIM + 32] };
        v2f b3 = { wp[48], wp[DIM + 48] };

        acc0 = __builtin_amdgcn_wmma_f32_16x16x4_f32(false, a, false, b0,
                                                     (short)0, acc0, false, false);
        acc1 = __builtin_amdgcn_wmma_f32_16x16x4_f32(false, a, false, b1,
                                                     (short)0, acc1, false, false);
        acc2 = __builtin_amdgcn_wmma_f32_16x16x4_f32(false, a, false, b2,
                                                     (short)0, acc2, false, false);
        acc3 = __builtin_amdgcn_wmma_f32_16x16x4_f32(false, a, false, b3,
                                                     (short)0, acc3, false, false);
    }

    // Store D: lane group selects rows r / r+8, col = lane%16 (+16*subtile)
    const int rbase = row0 + ((lane >> 4) << 3);
    const int cbase = n0 + (lane & 15);
#pragma unroll
    for (int r = 0; r < 8; ++r) {
        float* orow = O + (size_t)(rbase + r) * DIM + cbase;
        orow[ 0] = acc0[r];
        orow[16] = acc1[r];
        orow[32] = acc2[r];
        orow[48] = acc3[r];
    }
}

// ---------------------------------------------------------------------------
// Kernel 2: banded attention + output concat.
// One 256-thread block per (b,t) row.
// Reference semantics: invalid positions (pos<0) have key=value=0, so their
// logit is 0 (NOT -inf) and they DO participate in the softmax denominator.
// ---------------------------------------------------------------------------
__global__ __launch_bounds__(256) void attn_window(
    const float* __restrict__ X,
    const float* __restrict__ Q,
    const float* __restrict__ K,
    const float* __restrict__ V,
    float* __restrict__ out)
{
    const int row = blockIdx.x;            // 0..8191
    const int b   = row >> 10;
    const int t   = row & (TLEN - 1);
    const int tid = threadIdx.x;
    const int wave = tid >> 5;
    const int lane = tid & 31;

    __shared__ float s_q[DIM];
    __shared__ float s_logits[WIN];

    // stage q row in LDS (256 threads x 2)
    s_q[tid]       = Q[(size_t)row * DIM + tid];
    s_q[tid + 256] = Q[(size_t)row * DIM + tid + 256];
    __syncthreads();

    // each wave computes logits for w = wave and w = wave+8
#pragma unroll
    for (int wi = 0; wi < 2; ++wi) {
        const int w   = wave + 8 * wi;
        const int pos = t - (LEFTW - 1) + w;
        float acc = 0.0f;
        if (pos >= 0) {
            const float* kr = K + ((size_t)(b * TLEN + pos)) * DIM;
#pragma unroll 4
            for (int d = lane; d < DIM; d += 32)
                acc += s_q[d] * kr[d];
        }
#pragma unroll
        for (int off = 16; off > 0; off >>= 1)
            acc += __shfl_down(acc, off, 32);
        if (lane == 0)
            s_logits[w] = (pos >= 0) ? acc : 0.0f;   // zero-key logit = 0
    }
    __syncthreads();

    // softmax over 16 logits (redundant per thread, cheap)
    float mx = s_logits[0];
#pragma unroll
    for (int w = 1; w < WIN; ++w) mx = fmaxf(mx, s_logits[w]);
    float ex[WIN];
    float sum = 0.0f;
#pragma unroll
    for (int w = 0; w < WIN; ++w) {
        ex[w] = __expf(s_logits[w] - mx);
        sum += ex[w];
    }
    const float inv = 1.0f / sum;

    // weighted V sum + pass-through copy; each thread owns 2 columns
#pragma unroll
    for (int j = 0; j < 2; ++j) {
        const int d = tid + 256 * j;
        float a = 0.0f;
#pragma unroll
        for (int w = 0; w < WIN; ++w) {
            const int pos = t - (LEFTW - 1) + w;
            if (pos >= 0)
                a += ex[w] * V[((size_t)(b * TLEN + pos)) * DIM + d];
        }
        out[(size_t)row * (2 * DIM) + DIM + d] = a * inv;   // answer half
        out[(size_t)row * (2 * DIM) + d] = X[(size_t)row * DIM + d]; // input half
    }
}

extern "C" void kernel_launch(void* const* d_in, const int* in_sizes, int n_in,
                              void* d_out, int out_size, void* d_ws, size_t ws_size,
                              hipStream_t stream) {
    (void)in_sizes; (void)n_in; (void)out_size; (void)ws_size;

    const float* X  = (const float*)d_in[0];   // (8,1024,512)
    const float* Mw = (const float*)d_in[1];   // (512,512)
    const float* Cw = (const float*)d_in[2];   // (512,512)
    const float* Vw = (const float*)d_in[3];   // (512,512)
    float* out = (float*)d_out;                // (8,1024,1024)

    float* Q = (float*)d_ws;                   // 8192*512 f32
    float* K = Q + (size_t)NROWS * DIM;
    float* V = K + (size_t)NROWS * DIM;

    // 3 matrices * 4096 tiles, 8 waves per 256-thread block -> 1536 blocks
    qkv_gemm_wmma<<<1536, 256, 0, stream>>>(X, Mw, Cw, Vw, Q, K, V);

    // one block per (b,t) row
    attn_window<<<NROWS, 256, 0, stream>>>(X, Q, K, V, out);
}